// RPNPostProcessor_6622839571279
// MI455X (gfx1250) — compile-verified
//
#include <hip/hip_runtime.h>
#include <stdint.h>

// ---------------------------------------------------------------------------
// RPN post-processing for MI455X (gfx1250, wave32).
//   K1: radix-select top-2000 logits/image, bitonic sort, decode+clip boxes.
//   K2: NMS suppression bitmask build; pairwise area-sum on the matrix pipe
//       (V_WMMA_F32_16X16X4_F32 co-executes with the VALU min/max IoU work).
//   K3: single-wave greedy scan over the bitmask with TDM (tensor_load_to_lds)
//       double-buffering the mask rows into LDS; then prefix-sum compaction.
// ---------------------------------------------------------------------------

#define N_IMG 16
#define AN 3
#define HH 256
#define WW 336
#define HW (HH*WW)
#define NUM (AN*HW)          // 258048 anchors / image
#define KPAD 2048
#define PRE 2000
#define POST 1000
#define NMS_T 0.7f
#define NEGV (-1.0e30f)
#define BBOX_CLIP 4.135166556742356f   // log(1000/16)

typedef unsigned int u32;
typedef __attribute__((ext_vector_type(2))) float v2f;
typedef __attribute__((ext_vector_type(8))) float v8f;
typedef __attribute__((ext_vector_type(4))) u32  u32x4;
typedef __attribute__((ext_vector_type(4))) int  i32x4;
typedef __attribute__((ext_vector_type(8))) int  i32x8;

// monotone float -> u32 key (order-preserving; sigmoid is monotone so
// selecting on logits == selecting on scores)
__device__ __forceinline__ u32 fkey(float f){
  u32 u = __float_as_uint(f);
  return u ^ ((u32)((int)u >> 31) | 0x80000000u);
}
__device__ __forceinline__ float unfkey(u32 k){
  u32 u = (k & 0x80000000u) ? (k ^ 0x80000000u) : ~k;
  return __uint_as_float(u);
}

// wave32 lane broadcast via DS_BPERMUTE_B32
__device__ __forceinline__ float lane_bcast_f(float v, int srcLane){
  return __int_as_float(__builtin_amdgcn_ds_bpermute(srcLane << 2, __float_as_int(v)));
}

// ---------------------------------------------------------------------------
// TDM: 1D DMA of `ndw` dwords from global to LDS (contiguous).  D# per
// CDNA5 ISA.  Tracked on TENSORcnt.
// ---------------------------------------------------------------------------
__device__ __forceinline__ void tdm_load_dwords(u32 lds_off, const void* gsrc, u32 ndw){
  unsigned long long ga = (unsigned long long)(uintptr_t)gsrc;
  u32x4 g0;
  g0[0] = 1u;                                       // count=1, user descriptor
  g0[1] = lds_off;                                  // lds_addr (bytes)
  g0[2] = (u32)ga;                                  // global_addr[31:0]
  g0[3] = (u32)((ga >> 32) & 0x1FFFFFFu) | (2u << 30); // global_addr[56:32] | type=2
  i32x8 g1;
  g1[0] = (int)(2u << 16);                          // wg_mask=0, data_size=4B
  g1[1] = (int)((ndw & 0xFFFFu) << 16);             // tensor_dim0[15:0]
  g1[2] = (int)(((ndw >> 16) & 0xFFFFu) | (1u << 16)); // tensor_dim0[31:16], tensor_dim1=1
  g1[3] = (int)((ndw & 0xFFFFu) << 16);             // tile_dim0
  g1[4] = 0;                                        // tile_dim1/2 unused
  g1[5] = (int)ndw;                                 // tensor_dim0_stride lo
  g1[6] = 0;
  g1[7] = 0;
  i32x4 z4 = {0,0,0,0};
#if defined(__clang_major__) && (__clang_major__ >= 23)
  i32x8 z8 = {0,0,0,0,0,0,0,0};
  __builtin_amdgcn_tensor_load_to_lds(g0, g1, z4, z4, z8, 0);
#else
  __builtin_amdgcn_tensor_load_to_lds(g0, g1, z4, z4, 0);
#endif
}

// ---------------------------------------------------------------------------
// Kernel 1: per-image (16 blocks x 1024 threads).
// 3-level LDS radix-select on sortable keys -> exact top-2000 threshold,
// compaction, 2048-element bitonic sort in LDS, gather+decode+clip.
// ---------------------------------------------------------------------------
__global__ __launch_bounds__(1024) void k_select(
    const float* __restrict__ anchors, const float* __restrict__ objectness,
    const float* __restrict__ boxreg, const int* __restrict__ p_img_h,
    const int* __restrict__ p_img_w, float* __restrict__ wsBoxes,
    float* __restrict__ wsScores)
{
  __shared__ u32 hist[4096];
  __shared__ u32 coarse[1024];
  __shared__ u32 skey[KPAD];
  __shared__ u32 sidx[KPAD];
  __shared__ u32 sred[4];
  const int n   = blockIdx.x;
  const int tid = threadIdx.x;
  const float* __restrict__ obj = objectness + (size_t)n * NUM;

  // ---- level 1: top 12 bits ----
  for (int t = tid; t < 4096; t += 1024) hist[t] = 0u;
  __syncthreads();
  for (int i = tid; i < NUM; i += 1024){
    __builtin_prefetch(obj + i + 8192, 0, 0);        // global_prefetch_b8
    atomicAdd(&hist[fkey(obj[i]) >> 20], 1u);
  }
  __syncthreads();
  coarse[tid] = hist[4*tid] + hist[4*tid+1] + hist[4*tid+2] + hist[4*tid+3];
  __syncthreads();
  if (tid == 0){
    u32 acc = 0; int cb = 1023;
    for (; cb > 0; --cb){ if (acc + coarse[cb] >= PRE) break; acc += coarse[cb]; }
    int fb = 4*cb + 3;
    for (; fb > 4*cb; --fb){ if (acc + hist[fb] >= PRE) break; acc += hist[fb]; }
    sred[0] = (u32)fb; sred[1] = acc;
  }
  __syncthreads();
  const u32 b1 = sred[0]; const u32 acc1 = sred[1];
  __syncthreads();

  // ---- level 2: next 12 bits within threshold bin ----
  for (int t = tid; t < 4096; t += 1024) hist[t] = 0u;
  __syncthreads();
  for (int i = tid; i < NUM; i += 1024){
    u32 k = fkey(obj[i]);
    if ((k >> 20) == b1) atomicAdd(&hist[(k >> 8) & 0xFFFu], 1u);
  }
  __syncthreads();
  coarse[tid] = hist[4*tid] + hist[4*tid+1] + hist[4*tid+2] + hist[4*tid+3];
  __syncthreads();
  if (tid == 0){
    u32 acc = acc1; int cb = 1023;
    for (; cb > 0; --cb){ if (acc + coarse[cb] >= PRE) break; acc += coarse[cb]; }
    int fb = 4*cb + 3;
    for (; fb > 4*cb; --fb){ if (acc + hist[fb] >= PRE) break; acc += hist[fb]; }
    sred[0] = (u32)fb; sred[1] = acc;
  }
  __syncthreads();
  const u32 b2 = sred[0]; const u32 acc2 = sred[1];
  const u32 pref24 = (b1 << 12) | b2;
  __syncthreads();

  // ---- level 3: low 8 bits -> exact 32-bit threshold key ----
  for (int t = tid; t < 4096; t += 1024) hist[t] = 0u;
  __syncthreads();
  for (int i = tid; i < NUM; i += 1024){
    u32 k = fkey(obj[i]);
    if ((k >> 8) == pref24) atomicAdd(&hist[k & 0xFFu], 1u);
  }
  __syncthreads();
  if (tid == 0){
    u32 acc = acc2; int fb = 255;
    for (; fb > 0; --fb){ if (acc + hist[fb] >= PRE) break; acc += hist[fb]; }
    sred[0] = (u32)fb; sred[1] = acc; sred[2] = 0u; sred[3] = 0u;
  }
  __syncthreads();
  const u32 kthr  = (pref24 << 8) | sred[0];
  const u32 above = sred[1];                 // strictly greater, < PRE
  const u32 tieNeed = PRE - above;

  // ---- collect candidates (convert raw [A][H][W] index -> permuted (h*W+w)*A+a) ----
  for (int i = tid; i < NUM; i += 1024){
    u32 k = fkey(obj[i]);
    if (k < kthr) continue;
    int a = i / HW; int r = i - a*HW;
    u32 idx = (u32)(r * AN + a);
    if (k > kthr){
      u32 pos = atomicAdd(&sred[2], 1u);
      if (pos < PRE){ skey[pos] = k; sidx[pos] = idx; }
    } else {
      u32 t2 = atomicAdd(&sred[3], 1u);
      if (t2 < tieNeed){ u32 pos = above + t2; skey[pos] = k; sidx[pos] = idx; }
    }
  }
  for (int s = PRE + tid; s < KPAD; s += 1024){ skey[s] = 0u; sidx[s] = 0xFFFFFFFFu; }
  __syncthreads();

  // ---- bitonic sort 2048 (desc key, ties: asc idx == lax.top_k order) ----
  for (u32 ksz = 2; ksz <= KPAD; ksz <<= 1){
    for (u32 j = ksz >> 1; j > 0; j >>= 1){
      for (u32 e = tid; e < KPAD; e += 1024){
        u32 p = e ^ j;
        if (p > e){
          u32 ka = skey[e], kb = skey[p], ia = sidx[e], ib = sidx[p];
          bool aFirst = (ka > kb) || (ka == kb && ia < ib);
          bool desc = ((e & ksz) == 0);
          if (desc ? !aFirst : aFirst){
            skey[e] = kb; sidx[e] = ib; skey[p] = ka; sidx[p] = ia;
          }
        }
      }
      __syncthreads();
    }
  }

  // ---- gather + BoxCoder.decode + clip_to_image; score from inverted key ----
  const int img_h = *p_img_h, img_w = *p_img_w;
  const float fw = (float)(img_w - 1), fh = (float)(img_h - 1);
  const float* __restrict__ anc = anchors + (size_t)n * NUM * 4;
  const float* __restrict__ br  = boxreg  + (size_t)n * (AN*4) * HW;
  float* __restrict__ ob = wsBoxes  + (size_t)n * KPAD * 4;
  float* __restrict__ os = wsScores + (size_t)n * KPAD;
  for (int s = tid; s < KPAD; s += 1024){
    if (s < PRE){
      u32 k = skey[s]; u32 idx = sidx[s];
      int a = (int)(idx % AN); int r = (int)(idx / AN);
      float dx = br[(a*4+0)*HW + r];
      float dy = br[(a*4+1)*HW + r];
      float dw = fminf(br[(a*4+2)*HW + r], BBOX_CLIP);
      float dh = fminf(br[(a*4+3)*HW + r], BBOX_CLIP);
      float ax1 = anc[(size_t)idx*4+0], ay1 = anc[(size_t)idx*4+1];
      float ax2 = anc[(size_t)idx*4+2], ay2 = anc[(size_t)idx*4+3];
      float aw = ax2 - ax1 + 1.f, ah = ay2 - ay1 + 1.f;
      float cx = ax1 + 0.5f*aw,   cy = ay1 + 0.5f*ah;
      float pcx = dx*aw + cx,     pcy = dy*ah + cy;
      float pw = __expf(dw)*aw,   ph = __expf(dh)*ah;
      float x1 = fminf(fmaxf(pcx - 0.5f*pw,       0.f), fw);
      float y1 = fminf(fmaxf(pcy - 0.5f*ph,       0.f), fh);
      float x2 = fminf(fmaxf(pcx + 0.5f*pw - 1.f, 0.f), fw);
      float y2 = fminf(fmaxf(pcy + 0.5f*ph - 1.f, 0.f), fh);
      ob[s*4+0] = x1; ob[s*4+1] = y1; ob[s*4+2] = x2; ob[s*4+3] = y2;
      float logit = unfkey(k);
      os[s] = 1.f / (1.f + __expf(-logit));
    } else {
      ob[s*4+0] = 0.f; ob[s*4+1] = 0.f; ob[s*4+2] = 0.f; ob[s*4+3] = 0.f;
      os[s] = NEGV;
    }
  }
}

// ---------------------------------------------------------------------------
// Kernel 2: suppression bitmask. One wave owns a 16-row x 32-col tile.
// Pairwise (area_i + area_j) via V_WMMA_F32_16X16X4_F32 (rank-2 product),
// IoU test = inter > T*(sum - inter) on VALU, packed via ballot_w32.
// mask[n][i][w] bit b: suppress j = 32w+b when row i is kept (only j > i).
// ---------------------------------------------------------------------------
__global__ __launch_bounds__(256) void k_mask(
    const float* __restrict__ wsBoxes, u32* __restrict__ mask)
{
  const int lane = threadIdx.x & 31;
  const int gw   = blockIdx.x * 8 + (threadIdx.x >> 5);
  const int n    = gw >> 13;           // 8192 tiles / image
  const int rem  = gw & 8191;
  const int rt   = rem >> 6;
  const int ct   = rem & 63;
  const int i0   = rt * 16;
  const int j0   = ct * 32;
  u32* mrow = mask + ((size_t)n * KPAD + i0) * 64 + ct;

  if (j0 + 31 <= i0){                  // strictly lower triangle: all zero
    if (lane < 16) mrow[(size_t)lane * 64] = 0u;
    return;                            // wave-uniform exit: EXEC stays full for WMMA waves
  }

  const float* __restrict__ bb = wsBoxes + (size_t)n * KPAD * 4;
  const int ri = i0 + (lane & 15);
  float rx1 = bb[ri*4+0], ry1 = bb[ri*4+1], rx2 = bb[ri*4+2], ry2 = bb[ri*4+3];
  float areaR = (rx2 - rx1 + 1.f) * (ry2 - ry1 + 1.f);
  const int cj = j0 + lane;
  float qx1 = bb[cj*4+0], qy1 = bb[cj*4+1], qx2 = bb[cj*4+2], qy2 = bb[cj*4+3];
  float areaC = (qx2 - qx1 + 1.f) * (qy2 - qy1 + 1.f);

  // A (16x4): row m = (areaR[m], 1, 0, 0). Layout: lanes 0-15 hold K=0,1;
  // lanes 16-31 hold K=2,3 (zero here).
  v2f avec;
  avec.x = (lane < 16) ? areaR : 0.f;
  avec.y = (lane < 16) ? 1.f   : 0.f;

  u32 accLo[8], accHi[8];
  #pragma unroll
  for (int v = 0; v < 8; ++v){ accLo[v] = 0u; accHi[v] = 0u; }

  #pragma unroll
  for (int s = 0; s < 2; ++s){
    // B (4x16): col c = (1, areaC[j0+16s+c], 0, 0)
    float ac = lane_bcast_f(areaC, s*16 + (lane & 15));
    v2f bvec;
    bvec.x = (lane < 16) ? 1.f : 0.f;
    bvec.y = (lane < 16) ? ac  : 0.f;
    v8f csum = {};
    csum = __builtin_amdgcn_wmma_f32_16x16x4_f32(
        false, avec, false, bvec, (short)0, csum, false, false);
    #pragma unroll
    for (int v = 0; v < 8; ++v){
      int rsel = v + ((lane >= 16) ? 8 : 0);        // D layout: M = v (+8 hi half)
      float tx1 = lane_bcast_f(rx1, rsel);
      float ty1 = lane_bcast_f(ry1, rsel);
      float tx2 = lane_bcast_f(rx2, rsel);
      float ty2 = lane_bcast_f(ry2, rsel);
      int csel = s*16 + (lane & 15);
      float ux1 = lane_bcast_f(qx1, csel);
      float uy1 = lane_bcast_f(qy1, csel);
      float ux2 = lane_bcast_f(qx2, csel);
      float uy2 = lane_bcast_f(qy2, csel);
      float iw = fmaxf(fminf(tx2, ux2) - fmaxf(tx1, ux1) + 1.f, 0.f);
      float ih = fmaxf(fminf(ty2, uy2) - fmaxf(ty1, uy1) + 1.f, 0.f);
      float inter = iw * ih;
      float uni = csum[v] - inter;                  // (area_i + area_j) from matrix pipe
      int i = i0 + rsel;
      int j = j0 + csel;
      bool sup = (j > i) && (inter > NMS_T * uni);  // iou > T  <=>  inter > T*union
      u32 b = __builtin_amdgcn_ballot_w32(sup);
      accLo[v] |= (b & 0xFFFFu) << (16*s);          // row i0+v
      accHi[v] |= (b >> 16)     << (16*s);          // row i0+v+8
    }
  }
  if (lane == 0){
    #pragma unroll
    for (int v = 0; v < 8; ++v){
      mrow[(size_t)v       * 64] = accLo[v];
      mrow[(size_t)(v + 8) * 64] = accHi[v];
    }
  }
}

// ---------------------------------------------------------------------------
// Kernel 3: greedy scan + compaction. One block / image (256 threads).
// Wave 0 scans serially; mask rows streamed into LDS by the Tensor Data
// Mover in 16KB chunks, double-buffered, synchronized with s_wait_tensorcnt.
// ---------------------------------------------------------------------------
#define CHUNK_ROWS 64
#define CHUNK_DW   (CHUNK_ROWS * 64)     // 4096 dwords = 16 KB

__global__ __launch_bounds__(256) void k_nms_out(
    const float* __restrict__ wsBoxes, const float* __restrict__ wsScores,
    const u32* __restrict__ mask, float* __restrict__ out)
{
  __shared__ u32 mbuf[2][CHUNK_DW];
  __shared__ u32 keepS[64];
  __shared__ int preS[65];
  const int n    = blockIdx.x;
  const int tid  = threadIdx.x;
  const int lane = tid & 31;

  if (tid < 64){
    u32 w;
    if (tid < (PRE/32))       w = 0xFFFFFFFFu;
    else if (tid == (PRE/32)) w = (1u << (PRE & 31)) - 1u;
    else                      w = 0u;
    keepS[tid] = w;
  }
  __syncthreads();

  if ((tid >> 5) == 0){   // wave 0 only: serial greedy scan
    const u32* mbase = mask + (size_t)n * KPAD * 64;
    tdm_load_dwords((u32)(uintptr_t)(void*)&mbuf[0][0], mbase, CHUNK_DW);
    volatile u32* vk = keepS;
    for (int c = 0; c < KPAD / CHUNK_ROWS; ++c){
      const int par = c & 1;
      if (c + 1 < KPAD / CHUNK_ROWS){
        tdm_load_dwords((u32)(uintptr_t)(void*)&mbuf[(c + 1) & 1][0],
                        mbase + (size_t)(c + 1) * CHUNK_DW, CHUNK_DW);
        __builtin_amdgcn_s_wait_tensorcnt(1);   // chunk c complete (in-order)
      } else {
        __builtin_amdgcn_s_wait_tensorcnt(0);
      }
      for (int r = 0; r < CHUNK_ROWS; ++r){
        const int i = c * CHUNK_ROWS + r;
        u32 kw = vk[i >> 5];
        if ((kw >> (i & 31)) & 1u){
          u32 m0 = mbuf[par][r*64 + 2*lane];
          u32 m1 = mbuf[par][r*64 + 2*lane + 1];
          vk[2*lane]     = vk[2*lane]     & ~m0;   // each lane owns 2 keep words
          vk[2*lane + 1] = vk[2*lane + 1] & ~m1;
        }
      }
    }
  }
  __syncthreads();

  if (tid == 0){
    int acc = 0;
    for (int w = 0; w < 64; ++w){ preS[w] = acc; acc += __popc(keepS[w]); }
    preS[64] = acc;
  }
  __syncthreads();

  // boxes sorted by score desc => top-1000 of masked == kept-in-order then
  // unkept-in-order (ties at NEG resolved by ascending index, matching top_k).
  const int keptTotal = preS[64];
  const float* __restrict__ ob = wsBoxes  + (size_t)n * KPAD * 4;
  const float* __restrict__ os = wsScores + (size_t)n * KPAD;
  float* __restrict__ op = out + (size_t)n * POST * 5;
  for (int s = tid; s < KPAD; s += 256){
    u32 w = keepS[s >> 5];
    int sb = preS[s >> 5] + __popc(w & ((1u << (s & 31)) - 1u));
    bool kept = (w >> (s & 31)) & 1u;
    int pos = kept ? sb : (keptTotal + (s - sb));
    if (pos < POST){
      op[pos*5+0] = ob[s*4+0];
      op[pos*5+1] = ob[s*4+1];
      op[pos*5+2] = ob[s*4+2];
      op[pos*5+3] = ob[s*4+3];
      op[pos*5+4] = kept ? os[s] : NEGV;
    }
  }
}

// ---------------------------------------------------------------------------
extern "C" void kernel_launch(void* const* d_in, const int* in_sizes, int n_in,
                              void* d_out, int out_size, void* d_ws, size_t ws_size,
                              hipStream_t stream)
{
  (void)in_sizes; (void)n_in; (void)out_size; (void)ws_size;
  const float* anchors    = (const float*)d_in[0];
  const float* objectness = (const float*)d_in[1];
  const float* boxreg     = (const float*)d_in[2];
  const int*   img_h      = (const int*)d_in[3];
  const int*   img_w      = (const int*)d_in[4];
  float* out = (float*)d_out;

  char* ws = (char*)d_ws;
  const size_t boxesBytes  = (size_t)N_IMG * KPAD * 4 * sizeof(float);   // 512 KB
  const size_t scoresBytes = (size_t)N_IMG * KPAD * sizeof(float);       // 128 KB
  float* wsBoxes  = (float*)(ws);
  float* wsScores = (float*)(ws + boxesBytes);
  u32*   maskBuf  = (u32*)(ws + boxesBytes + scoresBytes);               // 8 MB

  k_select<<<N_IMG, 1024, 0, stream>>>(anchors, objectness, boxreg,
                                       img_h, img_w, wsBoxes, wsScores);

  const int nTiles  = N_IMG * (KPAD/16) * (KPAD/32);   // 131072 waves
  const int nBlocks = nTiles / 8;                       // 8 waves / block
  k_mask<<<nBlocks, 256, 0, stream>>>(wsBoxes, maskBuf);

  k_nms_out<<<N_IMG, 256, 0, stream>>>(wsBoxes, wsScores, maskBuf, out);
}